// GlobalNormCRFRLE_52578989637855
// MI455X (gfx1250) — compile-verified
//
#include <hip/hip_runtime.h>
#include <cmath>

typedef __attribute__((ext_vector_type(2))) float v2f;
typedef __attribute__((ext_vector_type(8))) float v8f;

#define NTIME   2000
#define NBATCH  128
#define INSIZE  256
#define NOUT    40
#define NEG_BIG (-1.0e30f)

#define NCHUNK  100
#define CLEN    (NTIME / NCHUNK)   // 20

// Branchless stable softplus: max(v,0) + log(1 + exp(-|v|)).
// exp arg <= 0 so e in (0,1], log arg in (1,2] where v_log_f32 is accurate.
__device__ __forceinline__ float softplusf(float v) {
    return fmaxf(v, 0.0f) + __logf(1.0f + __expf(-fabsf(v)));
}

// Branchless tanh via hardware exp + rcp: tanh(x) = sign(x)*(1 - 2*rcp(exp(2|x|)+1)).
// For large |x|, exp -> inf -> rcp -> 0 -> result -> 1 (correct saturation).
__device__ __forceinline__ float tanhf_fast(float v) {
    const float a = fabsf(v);
    const float e = __expf(2.0f * a);
    const float t = 1.0f - 2.0f * __builtin_amdgcn_rcpf(e + 1.0f);
    return copysignf(t, v);
}

__device__ __forceinline__ float lse8(const float (&t)[8]) {
    float mx = t[0];
#pragma unroll
    for (int m = 1; m < 8; ++m) mx = fmaxf(mx, t[m]);
    float s = 0.0f;
#pragma unroll
    for (int m = 0; m < 8; ++m) s += __expf(t[m] - mx);
    return mx + __logf(s);
}

__device__ __forceinline__ int iminc(int a, int b) { return a < b ? a : b; }

// ---------------------------------------------------------------------------
// Pass 1: y = x @ W^T + b via fp32 WMMA (16x16x4), fused activations.
// Each wave: 4 M-tiles (64 rows) x 3 N-tiles (48 cols, 40 valid).
// Block = 8 waves = 512 rows. Grid = 256000/512 = 500.
// ---------------------------------------------------------------------------
__global__ __launch_bounds__(256) void gemm_act_kernel(
    const float* __restrict__ x, const float* __restrict__ W,
    const float* __restrict__ bias, float* __restrict__ out)
{
    const int lane  = threadIdx.x & 31;
    const int wave  = threadIdx.x >> 5;
    const int l16   = lane & 15;
    const int halfk = lane >> 4;            // 0: K pair {0,1}; 1: K pair {2,3}

    const int rowBase = (blockIdx.x * 8 + wave) * 64;

    v8f acc[4][3];
#pragma unroll
    for (int m = 0; m < 4; ++m)
#pragma unroll
        for (int n = 0; n < 3; ++n)
#pragma unroll
            for (int i = 0; i < 8; ++i) acc[m][n][i] = 0.0f;

    // A fragment addresses: row = rowBase + mtile*16 + l16, cols kk+2*halfk..+1
    const float* xp = x + (size_t)(rowBase + l16) * INSIZE + 2 * halfk;
    // B fragment addresses: B[k][n] = W[n][k]; n clamped to 39 (junk cols never stored)
    const float* wp0 = W + (size_t)iminc(0 * 16 + l16, NOUT - 1) * INSIZE + 2 * halfk;
    const float* wp1 = W + (size_t)iminc(1 * 16 + l16, NOUT - 1) * INSIZE + 2 * halfk;
    const float* wp2 = W + (size_t)iminc(2 * 16 + l16, NOUT - 1) * INSIZE + 2 * halfk;

#pragma unroll 2
    for (int kk = 0; kk < INSIZE; kk += 4) {
        // x is streamed once: non-temporal so L2 keeps `out` for passes 2-4
        v2f a0 = __builtin_nontemporal_load((const v2f*)(xp + kk));
        v2f a1 = __builtin_nontemporal_load((const v2f*)(xp + 16 * INSIZE + kk));
        v2f a2 = __builtin_nontemporal_load((const v2f*)(xp + 32 * INSIZE + kk));
        v2f a3 = __builtin_nontemporal_load((const v2f*)(xp + 48 * INSIZE + kk));
        v2f b0 = *(const v2f*)(wp0 + kk);
        v2f b1 = *(const v2f*)(wp1 + kk);
        v2f b2 = *(const v2f*)(wp2 + kk);

        acc[0][0] = __builtin_amdgcn_wmma_f32_16x16x4_f32(false, a0, false, b0, (short)0, acc[0][0], false, false);
        acc[0][1] = __builtin_amdgcn_wmma_f32_16x16x4_f32(false, a0, false, b1, (short)0, acc[0][1], false, false);
        acc[0][2] = __builtin_amdgcn_wmma_f32_16x16x4_f32(false, a0, false, b2, (short)0, acc[0][2], false, false);
        acc[1][0] = __builtin_amdgcn_wmma_f32_16x16x4_f32(false, a1, false, b0, (short)0, acc[1][0], false, false);
        acc[1][1] = __builtin_amdgcn_wmma_f32_16x16x4_f32(false, a1, false, b1, (short)0, acc[1][1], false, false);
        acc[1][2] = __builtin_amdgcn_wmma_f32_16x16x4_f32(false, a1, false, b2, (short)0, acc[1][2], false, false);
        acc[2][0] = __builtin_amdgcn_wmma_f32_16x16x4_f32(false, a2, false, b0, (short)0, acc[2][0], false, false);
        acc[2][1] = __builtin_amdgcn_wmma_f32_16x16x4_f32(false, a2, false, b1, (short)0, acc[2][1], false, false);
        acc[2][2] = __builtin_amdgcn_wmma_f32_16x16x4_f32(false, a2, false, b2, (short)0, acc[2][2], false, false);
        acc[3][0] = __builtin_amdgcn_wmma_f32_16x16x4_f32(false, a3, false, b0, (short)0, acc[3][0], false, false);
        acc[3][1] = __builtin_amdgcn_wmma_f32_16x16x4_f32(false, a3, false, b1, (short)0, acc[3][1], false, false);
        acc[3][2] = __builtin_amdgcn_wmma_f32_16x16x4_f32(false, a3, false, b2, (short)0, acc[3][2], false, false);
    }

    // Epilogue: D layout = VGPR r: lanes0-15 -> M=r, lanes16-31 -> M=r+8; N=lane&15.
    // nt is a compile-time constant: nt==0 mixes softplus/tanh lanes -> branchless
    // select; nt>=1 is pure tanh (cols >= 16), no compares, no divergence.
#pragma unroll
    for (int m = 0; m < 4; ++m) {
#pragma unroll
        for (int nt = 0; nt < 3; ++nt) {
            const int col = nt * 16 + l16;
            if (col >= NOUT) continue;       // prunes only nt==2 lanes with l16 >= 8
            const float bb = bias[col];
            const int row0 = rowBase + m * 16 + halfk * 8;
#pragma unroll
            for (int r = 0; r < 8; ++r) {
                const float v = acc[m][nt][r] + bb;
                float res;
                if (nt == 0) {
                    const float sp   = softplusf(v);
                    const float th   = 5.0f * tanhf_fast(v);
                    const float ofs  = (col < 4) ? 1.0f : 0.1f;   // v_cndmask
                    res = (col < 8) ? (ofs + sp) : th;            // v_cndmask
                } else {
                    res = 5.0f * tanhf_fast(v);
                }
                out[(size_t)(row0 + r) * NOUT + col] = res;
            }
        }
    }
}

// ---------------------------------------------------------------------------
// Pass 2: per (batch, chunk) wave composes its 8x8 log-semiring matrices:
//   A_t[m][s] = w[i(s)][m] + mask(m,s),  M <- M (logmatmulexp) A_t
// 64 entries, 2 per lane. Masks use -1e30 sentinel (matches reference).
// ---------------------------------------------------------------------------
__global__ __launch_bounds__(32) void chunk_scan_kernel(
    const float* __restrict__ out, float* __restrict__ chunkM)
{
    __shared__ float Mld[64];
    __shared__ float wld[32];

    const int lane = threadIdx.x;
    const int b    = blockIdx.x & (NBATCH - 1);
    const int c    = blockIdx.x >> 7;
    const int s    = lane & 7;      // output-state column
    const int r0   = lane >> 3;     // rows r0 and r0+4
    const int irow = s & 3;         // w row index i(s)

    float mask[8];
#pragma unroll
    for (int m = 0; m < 8; ++m) {
        const bool on = (s < 4) ? (m == s || m == s + 4) : (m == s - 4 || m == s);
        mask[m] = (s < 4) ? (on ? NEG_BIG : 0.0f) : (on ? 0.0f : NEG_BIG);
    }

    const int t0 = c * CLEN;

    // step 0: M = A_{t0}
    wld[lane] = out[((size_t)t0 * NBATCH + b) * NOUT + 8 + lane];
    __syncthreads();
    float new0 = wld[irow * 8 + r0]     + mask[r0];
    float new1 = wld[irow * 8 + r0 + 4] + mask[r0 + 4];
    __syncthreads();
    Mld[r0 * 8 + s]       = new0;
    Mld[(r0 + 4) * 8 + s] = new1;

    for (int step = 1; step < CLEN; ++step) {
        const int t = t0 + step;
        const float wv = out[((size_t)t * NBATCH + b) * NOUT + 8 + lane];
        __syncthreads();                 // prior wld/Mld reads done
        wld[lane] = wv;
        __syncthreads();

        float A[8];
#pragma unroll
        for (int m = 0; m < 8; ++m) A[m] = wld[irow * 8 + m] + mask[m];

        float t0r[8], t1r[8];
#pragma unroll
        for (int m = 0; m < 8; ++m) {
            t0r[m] = Mld[r0 * 8 + m]       + A[m];
            t1r[m] = Mld[(r0 + 4) * 8 + m] + A[m];
        }
        new0 = lse8(t0r);
        new1 = lse8(t1r);
        __syncthreads();                 // all Mld reads done
        Mld[r0 * 8 + s]       = new0;
        Mld[(r0 + 4) * 8 + s] = new1;
    }

    float* dst = chunkM + ((size_t)b * NCHUNK + c) * 64;
    dst[r0 * 8 + s]       = new0;
    dst[(r0 + 4) * 8 + s] = new1;
}

// ---------------------------------------------------------------------------
// Pass 3: per-batch fold of chunk matrices: v_k <- lse_j(v_j + Mc[j][k]),
// starting v_k = lse_j(M0[j][k]) (fwd0 = zeros). logZdiv[b] = lse(v)/NTIME.
// ---------------------------------------------------------------------------
__global__ __launch_bounds__(32) void combine_kernel(
    const float* __restrict__ chunkM, float* __restrict__ logZdiv)
{
    __shared__ float Msh[64];
    __shared__ float vsh[8];
    const int lane = threadIdx.x;
    const int b    = blockIdx.x;
    const float* base = chunkM + (size_t)b * NCHUNK * 64;

    Msh[lane]      = base[lane];
    Msh[lane + 32] = base[lane + 32];
    __syncthreads();
    if (lane < 8) {
        float t[8];
#pragma unroll
        for (int j = 0; j < 8; ++j) t[j] = Msh[j * 8 + lane];
        vsh[lane] = lse8(t);
    }
    for (int c = 1; c < NCHUNK; ++c) {
        __syncthreads();
        Msh[lane]      = base[c * 64 + lane];
        Msh[lane + 32] = base[c * 64 + 32 + lane];
        __syncthreads();
        if (lane < 8) {
            float vold[8];
#pragma unroll
            for (int j = 0; j < 8; ++j) vold[j] = vsh[j];
            float t[8];
#pragma unroll
            for (int j = 0; j < 8; ++j) t[j] = vold[j] + Msh[j * 8 + lane];
            vsh[lane] = lse8(t);
        }
    }
    __syncthreads();
    if (lane == 0) {
        float t[8];
#pragma unroll
        for (int j = 0; j < 8; ++j) t[j] = vsh[j];
        logZdiv[b] = lse8(t) / (float)NTIME;
    }
}

// ---------------------------------------------------------------------------
// Fallback: single-workgroup sequential scan (used only if ws is tiny).
// 1024 threads = 128 batches x 8 states.
// ---------------------------------------------------------------------------
__global__ __launch_bounds__(1024) void seq_scan_kernel(
    const float* __restrict__ out, float* __restrict__ logZdiv)
{
    __shared__ float fwdsh[NBATCH * 8];
    const int tid  = threadIdx.x;
    const int b    = tid >> 3;
    const int s    = tid & 7;
    const int irow = s & 3;

    float mask[8];
#pragma unroll
    for (int m = 0; m < 8; ++m) {
        const bool on = (s < 4) ? (m == s || m == s + 4) : (m == s - 4 || m == s);
        mask[m] = (s < 4) ? (on ? NEG_BIG : 0.0f) : (on ? 0.0f : NEG_BIG);
    }

    fwdsh[tid] = 0.0f;
    __syncthreads();
    for (int t = 0; t < NTIME; ++t) {
        const float* wrow = out + ((size_t)t * NBATCH + b) * NOUT + 8 + irow * 8;
        const float4 w0 = *(const float4*)wrow;
        const float4 w1 = *(const float4*)(wrow + 4);
        float tt[8] = { w0.x + mask[0], w0.y + mask[1], w0.z + mask[2], w0.w + mask[3],
                        w1.x + mask[4], w1.y + mask[5], w1.z + mask[6], w1.w + mask[7] };
#pragma unroll
        for (int m = 0; m < 8; ++m) tt[m] += fwdsh[(b << 3) + m];
        const float nf = lse8(tt);
        __syncthreads();
        fwdsh[tid] = nf;
        __syncthreads();
    }
    if (s == 0) {
        float t[8];
#pragma unroll
        for (int j = 0; j < 8; ++j) t[j] = fwdsh[(b << 3) + j];
        logZdiv[b] = lse8(t) / (float)NTIME;
    }
}

// ---------------------------------------------------------------------------
// Pass 4: trans -= logZ[b]/NTIME (columns 8..39)
// ---------------------------------------------------------------------------
__global__ __launch_bounds__(256) void sublogz_kernel(
    float* __restrict__ out, const float* __restrict__ logZdiv)
{
    const size_t total = (size_t)NTIME * NBATCH * 32;
    const size_t e = (size_t)blockIdx.x * blockDim.x + threadIdx.x;
    if (e >= total) return;
    const int    j  = (int)(e & 31);
    const size_t tb = e >> 5;                 // t*128 + b
    const int    b  = (int)(tb & (NBATCH - 1));
    out[tb * NOUT + 8 + j] -= logZdiv[b];
}

extern "C" void kernel_launch(void* const* d_in, const int* in_sizes, int n_in,
                              void* d_out, int out_size, void* d_ws, size_t ws_size,
                              hipStream_t stream) {
    (void)in_sizes; (void)n_in; (void)out_size;
    const float* x    = (const float*)d_in[0];
    const float* W    = (const float*)d_in[1];
    const float* bias = (const float*)d_in[2];
    float* out = (float*)d_out;

    const int ROWS = NTIME * NBATCH;           // 256000
    gemm_act_kernel<<<ROWS / 512, 256, 0, stream>>>(x, W, bias, out);

    const size_t chunkBytes = (size_t)NBATCH * NCHUNK * 64 * sizeof(float);
    const size_t logZBytes  = (size_t)NBATCH * sizeof(float);
    const size_t total = (size_t)NTIME * NBATCH * 32;
    const int subBlocks = (int)((total + 255) / 256);

    if (ws_size >= chunkBytes + logZBytes) {
        float* chunkM  = (float*)d_ws;
        float* logZdiv = (float*)((char*)d_ws + chunkBytes);
        chunk_scan_kernel<<<NBATCH * NCHUNK, 32, 0, stream>>>(out, chunkM);
        combine_kernel<<<NBATCH, 32, 0, stream>>>(chunkM, logZdiv);
        sublogz_kernel<<<subBlocks, 256, 0, stream>>>(out, logZdiv);
    } else {
        float* logZdiv = (float*)d_ws;
        seq_scan_kernel<<<1, 1024, 0, stream>>>(out, logZdiv);
        sublogz_kernel<<<subBlocks, 256, 0, stream>>>(out, logZdiv);
    }
}